// FourierProbAttention_26963804684338
// MI455X (gfx1250) — compile-verified
//
#include <hip/hip_runtime.h>
#include <hip/hip_bf16.h>

#define BB 8
#define LL 2048
#define HH 8
#define DD 64
#define EE 33
#define EP 48            // E padded to 3 x 16 WMMA tiles
#define UU 40
#define UP 48            // top-k rows padded to 3 x 16 tiles
#define SK 40
#define ROWS (BB*HH*LL)  // 131072 (b,h,l)-rows
#define BH   (BB*HH)     // 64
#define PI_F 3.14159265358979323846f

#define KCH 8                 // split-K chunks for upd GEMM
#define KC_LEN (LL/KCH)       // 256 k-values per chunk
#define SUB 16                // k-rows staged per LDS buffer
#define NSUB (KC_LEN/SUB)     // 16 stages per chunk
#define TW_ENT (2*3*16*32)    // twiddle fragment entries (v2f each)

#if defined(__has_builtin)
#if __has_builtin(__builtin_amdgcn_global_load_async_to_lds_b128) && \
    __has_builtin(__builtin_amdgcn_s_wait_asynccnt)
#define USE_ASYNC_LDS 1
#endif
#endif
#ifndef USE_ASYNC_LDS
#define USE_ASYNC_LDS 0
#endif

typedef __attribute__((ext_vector_type(2))) float v2f;
typedef __attribute__((ext_vector_type(8))) float v8f;
typedef __attribute__((ext_vector_type(4))) int   v4i;

static __device__ __forceinline__ v8f wmma_f32(v2f a, v2f b, v8f c) {
  // V_WMMA_F32_16X16X4_F32 : D(16x16 f32) = A(16x4 f32) * B(4x16 f32) + C
  return __builtin_amdgcn_wmma_f32_16x16x4_f32(false, a, false, b, (short)0, c,
                                               false, false);
}

// ---------------------------------------------------------------------------
// K0: build DFT twiddle B-fragments once: tw[plane(2)][nt(3)][kt(16)][lane(32)]
// Each entry is the v2f a wave lane feeds into V_WMMA_F32_16X16X4_F32 as B.
// ---------------------------------------------------------------------------
__global__ __launch_bounds__(256)
void twiddle_init_kernel(float* __restrict__ tw) {
  const int t = blockIdx.x * 256 + threadIdx.x;
  if (t >= TW_ENT) return;
  const int plane = t / (3 * 16 * 32);
  const int rem   = t % (3 * 16 * 32);
  const int nt    = rem / (16 * 32);
  const int kt    = (rem / 32) % 16;
  const int lane  = rem % 32;
  const int lm = lane & 15;
  const int hi = lane >> 4;
  const int n  = nt * 16 + lm;
  const int kk = 4 * kt + 2 * hi;
  float vx = 0.f, vy = 0.f;
  if (n <= 32) {
    const int kn0 = (kk * n) & 63;
    const int kn1 = ((kk + 1) * n) & 63;
    float s0, c0, s1, c1;
    __sincosf((float)kn0 * (PI_F / 32.f), &s0, &c0);
    __sincosf((float)kn1 * (PI_F / 32.f), &s1, &c1);
    vx = plane ? -s0 : c0;   // plane 0: cos (Re), plane 1: -sin (Im)
    vy = plane ? -s1 : c1;
  }
  tw[2 * t]     = vx;
  tw[2 * t + 1] = vy;
}

// ---------------------------------------------------------------------------
// K1: rFFT of 16 rows as WMMA GEMM: X(16x64) x W(64x48pad) -> (Re, Im) planes.
// Twiddle B-fragments come from the precomputed table (resident in WGP$/L2).
// ---------------------------------------------------------------------------
__global__ __launch_bounds__(32)
void rfft16_kernel(const float* __restrict__ x, const float* __restrict__ tw,
                   float* __restrict__ oR, float* __restrict__ oI) {
  const int tile = blockIdx.x;       // ROWS/16 tiles
  const int lane = threadIdx.x;
  const int lm = lane & 15;
  const int hi = lane >> 4;
  const int r0 = tile * 16;
  const int r  = r0 + lm;
  const int b  = r >> 14;            // / (H*L)
  const int h  = (r >> 11) & (HH - 1);
  const int l  = r & (LL - 1);
  const float* xrow = x + (((size_t)(b * LL + l) * HH + h) * DD);

  const v2f* twR = (const v2f*)tw;          // plane 0
  const v2f* twI = twR + 3 * 16 * 32;       // plane 1

  v2f af[16];
#pragma unroll
  for (int kt = 0; kt < 16; ++kt) {
    const int kk = 4 * kt + 2 * hi;
    af[kt].x = xrow[kk];
    af[kt].y = xrow[kk + 1];
  }

#pragma unroll
  for (int nt = 0; nt < 3; ++nt) {
    v8f accR = {};
    v8f accI = {};
#pragma unroll
    for (int kt = 0; kt < 16; ++kt) {
      const v2f bR = twR[(nt * 16 + kt) * 32 + lane];
      const v2f bI = twI[(nt * 16 + kt) * 32 + lane];
      accR = wmma_f32(af[kt], bR, accR);
      accI = wmma_f32(af[kt], bI, accI);
    }
#pragma unroll
    for (int j = 0; j < 8; ++j) {
      const int row = r0 + j + hi * 8;
      const int col = nt * 16 + lm;
      oR[(size_t)row * EP + col] = accR[j];
      oI[(size_t)row * EP + col] = accI[j];
    }
  }
}

// ---------------------------------------------------------------------------
// K2: sampled QK and sparsity measure M (gather -> VALU dot products)
// ---------------------------------------------------------------------------
__global__ __launch_bounds__(256)
void sample_measure_kernel(const float* __restrict__ qr, const float* __restrict__ qi,
                           const float* __restrict__ kr, const float* __restrict__ ki,
                           const int* __restrict__ idx, float* __restrict__ Mout) {
  const int r = blockIdx.x * 256 + threadIdx.x;
  if (r >= ROWS) return;
  const int bh = r >> 11;
  const int l  = r & (LL - 1);
  const float* qrp = qr + (size_t)r * EP;
  const float* qip = qi + (size_t)r * EP;
  float maxR = -INFINITY, maxI = -INFINITY, sumR = 0.f, sumI = 0.f;
  for (int s = 0; s < SK; ++s) {
    const int ks = idx[l * SK + s];
    const float* krp = kr + (size_t)(bh * LL + ks) * EP;
    const float* kip = ki + (size_t)(bh * LL + ks) * EP;
    float aR = 0.f, aI = 0.f;
#pragma unroll 11
    for (int e = 0; e < EE; ++e) {
      const float qre = qrp[e], qie = qip[e];
      const float kre = krp[e], kie = kip[e];
      aR += qre * kre - qie * kie;   // complex mul, no conjugation (matches ref)
      aI += qre * kie + qie * kre;
    }
    maxR = fmaxf(maxR, aR); sumR += aR;
    maxI = fmaxf(maxI, aI); sumI += aI;
  }
  Mout[r] = maxR - sumR * (1.f / LL) + maxI - sumI * (1.f / LL);
}

// ---------------------------------------------------------------------------
// K3: top-40 of 2048 per (b,h) via iterative LDS argmax
// ---------------------------------------------------------------------------
__global__ __launch_bounds__(256)
void topk_kernel(const float* __restrict__ Min, int* __restrict__ Mtop) {
  __shared__ float vals[LL];
  __shared__ float redv[256];
  __shared__ int   redi[256];
  const int bh = blockIdx.x;
  const int t  = threadIdx.x;
  for (int i = t; i < LL; i += 256) vals[i] = Min[(size_t)bh * LL + i];
  __syncthreads();
  for (int u = 0; u < UU; ++u) {
    float bv = -INFINITY; int bi = 0;
    for (int i = t; i < LL; i += 256) {
      const float v = vals[i];
      if (v > bv) { bv = v; bi = i; }
    }
    redv[t] = bv; redi[t] = bi;
    __syncthreads();
    for (int off = 128; off > 0; off >>= 1) {
      if (t < off && redv[t + off] > redv[t]) {
        redv[t] = redv[t + off]; redi[t] = redi[t + off];
      }
      __syncthreads();
    }
    if (t == 0) {
      Mtop[bh * UU + u] = redi[0];
      vals[redi[0]] = -INFINITY;
    }
    __syncthreads();
  }
}

// ---------------------------------------------------------------------------
// K4: scores = Q_red x K^T (complex) via 4 real WMMA chains, scale 1/sqrt(D)
// grid: (L/16 n-tiles, BH), block: 1 wave
// ---------------------------------------------------------------------------
__global__ __launch_bounds__(32)
void scores_kernel(const float* __restrict__ qr, const float* __restrict__ qi,
                   const float* __restrict__ kr, const float* __restrict__ ki,
                   const int* __restrict__ Mtop,
                   float* __restrict__ sr, float* __restrict__ si) {
  const int nt   = blockIdx.x;
  const int bh   = blockIdx.y;
  const int lane = threadIdx.x;
  const int lm = lane & 15;
  const int hi = lane >> 4;

  int rowA[3]; bool valid[3];
#pragma unroll
  for (int mt = 0; mt < 3; ++mt) {
    const int u = mt * 16 + lm;
    valid[mt] = (u < UU);
    const int lsel = valid[mt] ? Mtop[bh * UU + u] : 0;
    rowA[mt] = bh * LL + lsel;
  }
  const int n    = nt * 16 + lm;
  const int rowK = bh * LL + n;

  const v8f z = {};
  v8f accR[3] = {z, z, z};
  v8f accI[3] = {z, z, z};

  for (int kt = 0; kt < EP / 4; ++kt) {
    const int kk = 4 * kt + 2 * hi;
    v2f bKr, bKi;
    bKr.x = kr[(size_t)rowK * EP + kk];  bKr.y = kr[(size_t)rowK * EP + kk + 1];
    bKi.x = ki[(size_t)rowK * EP + kk];  bKi.y = ki[(size_t)rowK * EP + kk + 1];
#pragma unroll
    for (int mt = 0; mt < 3; ++mt) {
      v2f aR = {0.f, 0.f}, aI = {0.f, 0.f};
      if (valid[mt]) {
        aR.x = qr[(size_t)rowA[mt] * EP + kk];
        aR.y = qr[(size_t)rowA[mt] * EP + kk + 1];
        aI.x = qi[(size_t)rowA[mt] * EP + kk];
        aI.y = qi[(size_t)rowA[mt] * EP + kk + 1];
      }
      const v2f aIn = {-aI.x, -aI.y};
      accR[mt] = wmma_f32(aR,  bKr, accR[mt]);   // + Qr*Kr
      accR[mt] = wmma_f32(aIn, bKi, accR[mt]);   // - Qi*Ki
      accI[mt] = wmma_f32(aR,  bKi, accI[mt]);   // + Qr*Ki
      accI[mt] = wmma_f32(aI,  bKr, accI[mt]);   // + Qi*Kr
    }
  }
  const float sc = 0.125f;  // 1/sqrt(64)
#pragma unroll
  for (int mt = 0; mt < 3; ++mt) {
#pragma unroll
    for (int j = 0; j < 8; ++j) {
      const int u = mt * 16 + j + hi * 8;
      sr[((size_t)bh * UP + u) * LL + n] = accR[mt][j] * sc;
      si[((size_t)bh * UP + u) * LL + n] = accI[mt][j] * sc;
    }
  }
}

// ---------------------------------------------------------------------------
// K5: in-place softmax over rows of length L (real & imag planes contiguous)
// ---------------------------------------------------------------------------
__global__ __launch_bounds__(256)
void softmax_kernel(float* __restrict__ base) {
  __shared__ float red[256];
  const int t = threadIdx.x;
  float* p = base + (size_t)blockIdx.x * LL;

  float m = -INFINITY;
  for (int i = t; i < LL; i += 256) m = fmaxf(m, p[i]);
  red[t] = m; __syncthreads();
  for (int off = 128; off > 0; off >>= 1) {
    if (t < off) red[t] = fmaxf(red[t], red[t + off]);
    __syncthreads();
  }
  const float mx = red[0];
  __syncthreads();

  float sum = 0.f;
  for (int i = t; i < LL; i += 256) {
    const float e = __expf(p[i] - mx);
    p[i] = e;
    sum += e;
  }
  red[t] = sum; __syncthreads();
  for (int off = 128; off > 0; off >>= 1) {
    if (t < off) red[t] += red[t + off];
    __syncthreads();
  }
  const float inv = 1.f / red[0];
  for (int i = t; i < LL; i += 256) p[i] *= inv;
}

// ---------------------------------------------------------------------------
// K6: upd partial GEMM: attn(48 x 256) x v(256 x 48) per (bh, plane, k-chunk).
// V tile double-buffered in LDS via async global->LDS when available; WMMA B
// fragments are then served by ds_load.
// ---------------------------------------------------------------------------
__global__ __launch_bounds__(32)
void upd_kernel(const float* __restrict__ attn_base, const float* __restrict__ v_base,
                float* __restrict__ upd_part) {
  const int bh    = blockIdx.x;
  const int plane = blockIdx.y;  // 0: real, 1: imag
  const int kc    = blockIdx.z;  // split-K chunk
  const float* A  = attn_base + (size_t)plane * BH * UP * LL + (size_t)bh * UP * LL;
  const float* Vp = v_base + (size_t)plane * ROWS * EP + (size_t)bh * LL * EP
                           + (size_t)kc * KC_LEN * EP;
  float* Up = upd_part + ((size_t)kc * 2 + plane) * BH * UP * EP + (size_t)bh * UP * EP;

  const int lane = threadIdx.x;
  const int lm = lane & 15;
  const int hi = lane >> 4;

  __shared__ __align__(16) float vst[2][SUB * EP];   // 2 x 3 KB V tiles

  const v8f z = {};
  v8f acc[3][3] = {{z, z, z}, {z, z, z}, {z, z, z}};

  auto stage = [&](int s, int buf) {
#pragma unroll
    for (int j = 0; j < 6; ++j) {                 // 192 float4 / 32 lanes
      const int t4 = lane + 32 * j;
      const float* src = Vp + (size_t)s * SUB * EP + (size_t)t4 * 4;
      float* dst = &vst[buf][t4 * 4];
#if USE_ASYNC_LDS
      __builtin_amdgcn_global_load_async_to_lds_b128((v4i*)src, (v4i*)dst, 0, 0);
#else
      *(float4*)dst = *(const float4*)src;
#endif
    }
  };

  stage(0, 0);
  for (int s = 0; s < NSUB; ++s) {
    if (s + 1 < NSUB) {
      stage(s + 1, (s + 1) & 1);
#if USE_ASYNC_LDS
      __builtin_amdgcn_s_wait_asynccnt(6);   // stage s complete; s+1 in flight
#endif
    } else {
#if USE_ASYNC_LDS
      __builtin_amdgcn_s_wait_asynccnt(0);
#endif
    }
    __syncthreads();
    const float* vb = vst[s & 1];
#pragma unroll
    for (int q = 0; q < SUB / 4; ++q) {           // 4 WMMA K-steps per stage
      const int klocal = q * 4 + 2 * hi;
      const int kglob  = kc * KC_LEN + s * SUB + q * 4 + 2 * hi;
      v2f a[3], bfr[3];
#pragma unroll
      for (int mt = 0; mt < 3; ++mt) {
        a[mt].x = A[(size_t)(mt * 16 + lm) * LL + kglob];
        a[mt].y = A[(size_t)(mt * 16 + lm) * LL + kglob + 1];
      }
#pragma unroll
      for (int nb = 0; nb < 3; ++nb) {
        bfr[nb].x = vb[klocal * EP + nb * 16 + lm];
        bfr[nb].y = vb[(klocal + 1) * EP + nb * 16 + lm];
      }
#pragma unroll
      for (int mt = 0; mt < 3; ++mt)
#pragma unroll
        for (int nb = 0; nb < 3; ++nb)
          acc[mt][nb] = wmma_f32(a[mt], bfr[nb], acc[mt][nb]);
    }
    __syncthreads();
  }
#pragma unroll
  for (int mt = 0; mt < 3; ++mt)
#pragma unroll
    for (int nb = 0; nb < 3; ++nb)
#pragma unroll
      for (int j = 0; j < 8; ++j)
        Up[(size_t)(mt * 16 + j + hi * 8) * EP + nb * 16 + lm] = acc[mt][nb][j];
}

// ---------------------------------------------------------------------------
// K7: deterministic reduction of the split-K partials
// ---------------------------------------------------------------------------
__global__ __launch_bounds__(256)
void upd_reduce_kernel(const float* __restrict__ part, float* __restrict__ ur) {
  const size_t N = (size_t)2 * BH * UP * EP;
  const size_t i = (size_t)blockIdx.x * 256 + threadIdx.x;
  if (i >= N) return;
  float s = 0.f;
#pragma unroll
  for (int kc = 0; kc < KCH; ++kc) s += part[(size_t)kc * N + i];
  ur[i] = s;
}

// ---------------------------------------------------------------------------
// K8: v mean over L (both planes)
// ---------------------------------------------------------------------------
__global__ __launch_bounds__(96)
void vmean_kernel(const float* __restrict__ v_base, float* __restrict__ vm) {
  const int bh = blockIdx.x;
  const int t  = threadIdx.x;           // 0..95
  const int plane = t / EP;
  const int e     = t % EP;
  const float* Vp = v_base + (size_t)plane * ROWS * EP + (size_t)bh * LL * EP + e;
  float s = 0.f;
  for (int l = 0; l < LL; ++l) s += Vp[(size_t)l * EP];
  vm[(size_t)plane * BH * EP + (size_t)bh * EP + e] = s * (1.f / LL);
}

// ---------------------------------------------------------------------------
// K9: irFFT of the per-(b,h) mean row -> ymean(BH x 64)
// ---------------------------------------------------------------------------
__global__ __launch_bounds__(64)
void ymean_kernel(const float* __restrict__ vm, float* __restrict__ ym) {
  const int bh = blockIdx.x;
  const int d  = threadIdx.x;
  const float* mr = vm + (size_t)bh * EP;
  const float* mi = vm + (size_t)BH * EP + (size_t)bh * EP;
  float acc = 0.f;
  for (int e = 0; e < EE; ++e) {
    const float w = (e == 0 || e == 32) ? 1.f : 2.f;
    const int kn = (e * d) & 63;
    float sn, cn;
    __sincosf((float)kn * (PI_F / 32.f), &sn, &cn);
    acc += w * (mr[e] * cn - mi[e] * sn);
  }
  ym[(size_t)bh * DD + d] = acc * (1.f / DD);
}

// ---------------------------------------------------------------------------
// K10: broadcast ymean to every row of the output (float4 streaming stores)
// ---------------------------------------------------------------------------
__global__ __launch_bounds__(256)
void bcast_kernel(const float* __restrict__ ym, float* __restrict__ out) {
  const size_t i4 = (size_t)blockIdx.x * 256 + threadIdx.x;   // float4 index
  if (i4 >= (size_t)ROWS * DD / 4) return;
  const float4* y4 = (const float4*)ym;
  ((float4*)out)[i4] = y4[(i4 >> 15) * 16 + (i4 & 15)];  // bh = i4/(L*D/4), d4 = i4&15
}

// ---------------------------------------------------------------------------
// K11: irFFT of the 40 updated rows, scattered to their top-k positions
// ---------------------------------------------------------------------------
__global__ __launch_bounds__(256)
void scatter_kernel(const float* __restrict__ upd_base, const int* __restrict__ Mtop,
                    float* __restrict__ out) {
  const int bh = blockIdx.x;
  for (int w = threadIdx.x; w < UU * DD; w += 256) {
    const int u = w / DD;
    const int d = w % DD;
    const int l = Mtop[bh * UU + u];
    const float* urow = upd_base + (size_t)bh * UP * EP + (size_t)u * EP;
    const float* irow = upd_base + (size_t)BH * UP * EP + (size_t)bh * UP * EP + (size_t)u * EP;
    float acc = 0.f;
    for (int e = 0; e < EE; ++e) {
      const float wgt = (e == 0 || e == 32) ? 1.f : 2.f;
      const int kn = (e * d) & 63;
      float sn, cn;
      __sincosf((float)kn * (PI_F / 32.f), &sn, &cn);
      acc += wgt * (urow[e] * cn - irow[e] * sn);
    }
    out[((size_t)bh * LL + l) * DD + d] = acc * (1.f / DD);
  }
}

// ---------------------------------------------------------------------------
extern "C" void kernel_launch(void* const* d_in, const int* in_sizes, int n_in,
                              void* d_out, int out_size, void* d_ws, size_t ws_size,
                              hipStream_t stream) {
  (void)in_sizes; (void)n_in; (void)out_size; (void)ws_size;
  const float* q   = (const float*)d_in[0];
  const float* k   = (const float*)d_in[1];
  const float* v   = (const float*)d_in[2];
  // d_in[3] = attn_mask scalar (unused)
  const int*   idx = (const int*)d_in[4];
  float* out = (float*)d_out;

  // Workspace layout (floats)
  float* ws   = (float*)d_ws;
  const size_t PL = (size_t)ROWS * EP;        // one rfft plane
  float* tw = ws;                              // 2*TW_ENT floats (twiddle frags)
  float* qr = tw + 2 * TW_ENT;
  float* qi = qr + PL;
  float* kr = qi + PL;
  float* ki = kr + PL;
  float* vr = ki + PL;
  float* vi = vr + PL;
  float* Mbuf = vi + PL;                       // ROWS
  int*   Mtop = (int*)(Mbuf + ROWS);           // BH*UU ints (padded to 4096 floats)
  float* sr   = (float*)(Mbuf + ROWS) + 4096;  // 2 planes x BH*UP*LL (attn in place)
  float* ur   = sr + 2 * (size_t)BH * UP * LL; // 2 planes x BH*UP*EP
  float* vm   = ur + 2 * (size_t)BH * UP * EP; // 2 planes x BH*EP
  float* ym   = vm + 2 * (size_t)BH * EP;      // BH*DD
  float* upp  = ym + (size_t)BH * DD;          // KCH x 2 x BH*UP*EP partials

  // 0) DFT twiddle fragment table
  twiddle_init_kernel<<<dim3((TW_ENT + 255) / 256), 256, 0, stream>>>(tw);

  // 1) rFFT planes via WMMA GEMMs
  rfft16_kernel<<<dim3(ROWS / 16), 32, 0, stream>>>(q, tw, qr, qi);
  rfft16_kernel<<<dim3(ROWS / 16), 32, 0, stream>>>(k, tw, kr, ki);
  rfft16_kernel<<<dim3(ROWS / 16), 32, 0, stream>>>(v, tw, vr, vi);

  // 2) sampled sparsity measure + top-k
  sample_measure_kernel<<<dim3(ROWS / 256), 256, 0, stream>>>(qr, qi, kr, ki, idx, Mbuf);
  topk_kernel<<<dim3(BH), 256, 0, stream>>>(Mbuf, Mtop);

  // 3) complex scores GEMM (gathered q rows) + softmax (real & imag planes)
  scores_kernel<<<dim3(LL / 16, BH), 32, 0, stream>>>(qr, qi, kr, ki, Mtop,
                                                      sr, sr + (size_t)BH * UP * LL);
  softmax_kernel<<<dim3(2 * BH * UP), 256, 0, stream>>>(sr);

  // 4) attn x v split-K GEMM (async-LDS staged V) + deterministic reduce
  upd_kernel<<<dim3(BH, 2, KCH), 32, 0, stream>>>(sr, vr, upp);
  upd_reduce_kernel<<<dim3((int)(((size_t)2 * BH * UP * EP + 255) / 256)), 256, 0,
                      stream>>>(upp, ur);

  // 5) context mean + irFFT + scatter
  vmean_kernel<<<dim3(BH), 96, 0, stream>>>(vr, vm);
  ymean_kernel<<<dim3(BH), 64, 0, stream>>>(vm, ym);
  bcast_kernel<<<dim3((int)(((size_t)ROWS * DD / 4 + 255) / 256)), 256, 0, stream>>>(ym, out);
  scatter_kernel<<<dim3(BH), 256, 0, stream>>>(ur, Mtop, out);
}